// GraphTransformer_34282428957167
// MI455X (gfx1250) — compile-verified
//
#include <hip/hip_runtime.h>
#include <math.h>

// ---------------------------------------------------------------------------
// Problem constants (match reference)
// ---------------------------------------------------------------------------
constexpr int kN    = 32768;
constexpr int kE    = 524288;
constexpr int kF    = 128;
constexpr int kD    = 128;
constexpr int kL    = 8;
constexpr int kH    = 8;
constexpr int kC    = 2048;
constexpr int kNCLS = 10;
constexpr int kMLP  = 512;
constexpr int kS    = kC;        // tokens after cluster pooling
constexpr int kTD   = 3 * kD;    // qkv row stride
constexpr int kEdges = kE + kN;  // edges + self loops
constexpr float kEPS = 1e-5f;

typedef float v2f __attribute__((ext_vector_type(2)));
typedef float v8f __attribute__((ext_vector_type(8)));

#define DEV static __device__ __forceinline__

// CDNA5 fp32 WMMA: D(16x16,f32) = A(16x4,f32) * B(4x16,f32) + C
DEV v8f wmma_f32(v2f a, v2f b, v8f c) {
  return __builtin_amdgcn_wmma_f32_16x16x4_f32(
      /*neg_a=*/false, a, /*neg_b=*/false, b,
      /*c_mod=*/(short)0, c, /*reuse_a=*/false, /*reuse_b=*/false);
}

// ---------------------------------------------------------------------------
// Generic WMMA GEMM: out[M,N] = act(A[M,K] @ W[K,N] + bias) + res
// One wave computes a 16(M) x 32(N) tile. K multiple of 4; N multiple of 32.
// act: 0 = none, 1 = exact GELU
// ---------------------------------------------------------------------------
__global__ __launch_bounds__(256) void gemm_kernel(
    const float* __restrict__ A, const float* __restrict__ W,
    const float* __restrict__ bias, const float* __restrict__ res,
    float* __restrict__ out, int M, int N, int K, int act) {
  const int lane = threadIdx.x & 31;
  const int warp = threadIdx.x >> 5;
  const int l16  = lane & 15;
  const int half = lane >> 4;
  const int tilesN = N >> 5;
  const int wg = blockIdx.x * 8 + warp;
  const int tm = wg / tilesN;
  const int tn = wg % tilesN;
  if (tm * 16 >= M) return;

  const int row = tm * 16 + l16;     // A-operand: lane holds row (lane%16)
  const int c0  = tn * 32 + l16;     // B/C-operand: lane holds column
  const int c1  = c0 + 16;

  v8f acc0 = {};
  v8f acc1 = {};
  const float* arow = A + (size_t)row * K + 2 * half;  // K-halves split by lane/16

  for (int k = 0; k < K; k += 4) {
    v2f a;
    const float2 av = *(const float2*)(arow + k);
    a.x = av.x; a.y = av.y;
    const int kr = k + 2 * half;
    v2f b0, b1;
    b0.x = W[(size_t)kr * N + c0];
    b0.y = W[(size_t)(kr + 1) * N + c0];
    b1.x = W[(size_t)kr * N + c1];
    b1.y = W[(size_t)(kr + 1) * N + c1];
    acc0 = wmma_f32(a, b0, acc0);
    acc1 = wmma_f32(a, b1, acc1);
  }

  const float bv0 = bias ? bias[c0] : 0.f;
  const float bv1 = bias ? bias[c1] : 0.f;
#pragma unroll
  for (int j = 0; j < 8; ++j) {
    const int r = tm * 16 + j + 8 * half;  // C layout: VGPR j -> row j (+8 for hi half)
    float v0 = acc0[j] + bv0;
    float v1 = acc1[j] + bv1;
    if (act == 1) {
      v0 = 0.5f * v0 * (1.f + erff(v0 * 0.70710678118f));
      v1 = 0.5f * v1 * (1.f + erff(v1 * 0.70710678118f));
    }
    const size_t i0 = (size_t)r * N + c0;
    const size_t i1 = (size_t)r * N + c1;
    if (res) { v0 += res[i0]; v1 += res[i1]; }
    out[i0] = v0;
    out[i1] = v1;
  }
}

// ---------------------------------------------------------------------------
// Flash attention over pooled tokens: S=2048, H=8, hd=16.
// One wave handles (head, 16-query block); online softmax; WMMA for QK^T and
// P*V; P staged via LDS (C-layout -> A-layout) with explicit s_wait_dscnt.
// ---------------------------------------------------------------------------
__global__ __launch_bounds__(256) void attn_kernel(const float* __restrict__ qkv,
                                                   float* __restrict__ outp) {
  __shared__ float lds[8 * 256];
  const int lane = threadIdx.x & 31;
  const int warp = threadIdx.x >> 5;
  const int l16  = lane & 15;
  const int half = lane >> 4;
  const int wg   = blockIdx.x * 8 + warp;
  const int head = wg >> 7;       // kS/16 == 128 query blocks per head
  const int qb   = wg & 127;
  float* P = lds + warp * 256;

  // Preload Q operand (16x16 tile, four K=4 chunks)
  const int qrow = qb * 16 + l16;
  v2f aq[4];
#pragma unroll
  for (int cc = 0; cc < 4; ++cc) {
    const float2 v = *(const float2*)
        &qkv[(size_t)qrow * kTD + head * 16 + cc * 4 + 2 * half];
    aq[cc].x = v.x; aq[cc].y = v.y;
  }

  v8f o = {};
  float mrow[8], lrow[8];
#pragma unroll
  for (int j = 0; j < 8; ++j) { mrow[j] = -INFINITY; lrow[j] = 0.f; }

  for (int kb = 0; kb < kS / 16; ++kb) {
    // S_tile = Q * K^T  (B operand: K[n][k] contiguous in head-dim)
    v8f sc = {};
    const size_t krow = (size_t)(kb * 16 + l16) * kTD + kD + head * 16;
#pragma unroll
    for (int cc = 0; cc < 4; ++cc) {
      const float2 v = *(const float2*)&qkv[krow + cc * 4 + 2 * half];
      v2f b; b.x = v.x; b.y = v.y;
      sc = wmma_f32(aq[cc], b, sc);
    }
    // Online softmax per row (row j+8*half lives across this lane-half)
#pragma unroll
    for (int j = 0; j < 8; ++j) {
      float s = sc[j] * 0.25f;  // 1/sqrt(hd=16)
      float rmax = s;
      rmax = fmaxf(rmax, __shfl_xor(rmax, 1, 16));
      rmax = fmaxf(rmax, __shfl_xor(rmax, 2, 16));
      rmax = fmaxf(rmax, __shfl_xor(rmax, 4, 16));
      rmax = fmaxf(rmax, __shfl_xor(rmax, 8, 16));
      const float nm   = fmaxf(mrow[j], rmax);
      const float corr = __expf(mrow[j] - nm);
      const float p    = __expf(s - nm);
      float rs = p;
      rs += __shfl_xor(rs, 1, 16);
      rs += __shfl_xor(rs, 2, 16);
      rs += __shfl_xor(rs, 4, 16);
      rs += __shfl_xor(rs, 8, 16);
      lrow[j] = lrow[j] * corr + rs;
      o[j] *= corr;
      mrow[j] = nm;
      P[(j + 8 * half) * 16 + l16] = p;  // C-layout -> row-major stash
    }
    asm volatile("s_wait_dscnt 0" ::: "memory");  // LDS RAW within wave
    // O += P * V   (A operand from LDS, B operand: V rows)
#pragma unroll
    for (int cc = 0; cc < 4; ++cc) {
      v2f a;
      a.x = P[l16 * 16 + cc * 4 + 2 * half];
      a.y = P[l16 * 16 + cc * 4 + 2 * half + 1];
      const int vr = kb * 16 + cc * 4 + 2 * half;
      v2f b;
      b.x = qkv[(size_t)vr * kTD + 2 * kD + head * 16 + l16];
      b.y = qkv[(size_t)(vr + 1) * kTD + 2 * kD + head * 16 + l16];
      o = wmma_f32(a, b, o);
    }
    asm volatile("s_wait_dscnt 0" ::: "memory");  // before next-iter P stores
  }

#pragma unroll
  for (int j = 0; j < 8; ++j) {
    const int r = qb * 16 + j + 8 * half;
    outp[(size_t)r * kD + head * 16 + l16] = o[j] / lrow[j];
  }
}

// ---------------------------------------------------------------------------
// LayerNorm over D=128; one wave per row.
// ---------------------------------------------------------------------------
__global__ __launch_bounds__(256) void ln_kernel(
    const float* __restrict__ x, const float* __restrict__ g,
    const float* __restrict__ b, float* __restrict__ y, int S) {
  const int lane = threadIdx.x & 31;
  const int row = blockIdx.x * 8 + (threadIdx.x >> 5);
  if (row >= S) return;
  const float4 v = *(const float4*)&x[(size_t)row * kD + lane * 4];
  float s = v.x + v.y + v.z + v.w;
  for (int m = 1; m < 32; m <<= 1) s += __shfl_xor(s, m, 32);
  const float mean = s * (1.f / kD);
  const float dx = v.x - mean, dy = v.y - mean, dz = v.z - mean, dw = v.w - mean;
  float vv = dx * dx + dy * dy + dz * dz + dw * dw;
  for (int m = 1; m < 32; m <<= 1) vv += __shfl_xor(vv, m, 32);
  const float rstd = rsqrtf(vv * (1.f / kD) + kEPS);
  const float4 gg = *(const float4*)&g[lane * 4];
  const float4 bb = *(const float4*)&b[lane * 4];
  float4 oo;
  oo.x = dx * rstd * gg.x + bb.x;
  oo.y = dy * rstd * gg.y + bb.y;
  oo.z = dz * rstd * gg.z + bb.z;
  oo.w = dw * rstd * gg.w + bb.w;
  *(float4*)&y[(size_t)row * kD + lane * 4] = oo;
}

// ---------------------------------------------------------------------------
// GAT helpers
// ---------------------------------------------------------------------------
DEV void edge_ids(int id, const int* ei, int& s, int& d) {
  if (id < kE) { s = ei[id]; d = ei[kE + id]; }
  else { s = id - kE; d = s; }  // self loops appended
}
DEV float leaky(float e) { return e > 0.f ? e : 0.2f * e; }
DEV unsigned f2ord(float f) {
  unsigned u = __float_as_uint(f);
  return (u & 0x80000000u) ? ~u : (u | 0x80000000u);
}
DEV float ord2f(unsigned u) {
  const unsigned v = (u & 0x80000000u) ? (u ^ 0x80000000u) : ~u;
  return __uint_as_float(v);
}

// es = h @ a_s, ed = h @ a_d ; one wave per node
__global__ __launch_bounds__(256) void gat_dots_kernel(
    const float* __restrict__ h, const float* __restrict__ as,
    const float* __restrict__ ad, float* __restrict__ es, float* __restrict__ ed) {
  const int lane = threadIdx.x & 31;
  const int n = blockIdx.x * 8 + (threadIdx.x >> 5);
  if (n >= kN) return;
  const float4 v = *(const float4*)&h[(size_t)n * kD + lane * 4];
  const float4 a = *(const float4*)&as[lane * 4];
  const float4 d = *(const float4*)&ad[lane * 4];
  float s = v.x * a.x + v.y * a.y + v.z * a.z + v.w * a.w;
  float t = v.x * d.x + v.y * d.y + v.z * d.z + v.w * d.w;
  for (int m = 1; m < 32; m <<= 1) {
    s += __shfl_xor(s, m, 32);
    t += __shfl_xor(t, m, 32);
  }
  if (lane == 0) { es[n] = s; ed[n] = t; }
}

// out[n,:] = bias ; menc[n] = 0 ; z[n] = 0
__global__ void gat_init_kernel(float* __restrict__ out,
                                const float* __restrict__ bias,
                                unsigned* __restrict__ menc,
                                float* __restrict__ z) {
  const size_t i = (size_t)blockIdx.x * blockDim.x + threadIdx.x;
  if (i >= (size_t)kN * kD) return;
  const int d = (int)(i & (kD - 1));
  out[i] = bias[d];
  if (d == 0) {
    const int n = (int)(i >> 7);
    menc[n] = 0u;
    z[n] = 0.f;
  }
}

__global__ void edge_max_kernel(const float* __restrict__ es,
                                const float* __restrict__ ed,
                                const int* __restrict__ ei,
                                unsigned* __restrict__ menc) {
  const int id = blockIdx.x * blockDim.x + threadIdx.x;
  if (id >= kEdges) return;
  int s, d; edge_ids(id, ei, s, d);
  const float e = leaky(es[s] + ed[d]);
  atomicMax(&menc[d], f2ord(e));
}

__global__ void edge_sum_kernel(const float* __restrict__ es,
                                const float* __restrict__ ed,
                                const int* __restrict__ ei,
                                const unsigned* __restrict__ menc,
                                float* __restrict__ z) {
  const int id = blockIdx.x * blockDim.x + threadIdx.x;
  if (id >= kEdges) return;
  int s, d; edge_ids(id, ei, s, d);
  const float e = leaky(es[s] + ed[d]);
  atomicAdd(&z[d], __expf(e - ord2f(menc[d])));
}

// out[dst] += alpha * h[src]; one wave per edge, 4 floats per lane
__global__ __launch_bounds__(256) void edge_aggr_kernel(
    const float* __restrict__ es, const float* __restrict__ ed,
    const int* __restrict__ ei, const unsigned* __restrict__ menc,
    const float* __restrict__ z, const float* __restrict__ h,
    float* __restrict__ out) {
  const int lane = threadIdx.x & 31;
  const int id = blockIdx.x * 8 + (threadIdx.x >> 5);
  if (id >= kEdges) return;
  int s, d; edge_ids(id, ei, s, d);
  const float e = leaky(es[s] + ed[d]);
  const float alpha = __expf(e - ord2f(menc[d])) / z[d];
  const float4 v = *(const float4*)&h[(size_t)s * kD + lane * 4];
  float* o = out + (size_t)d * kD + lane * 4;
  atomicAdd(o + 0, alpha * v.x);
  atomicAdd(o + 1, alpha * v.y);
  atomicAdd(o + 2, alpha * v.z);
  atomicAdd(o + 3, alpha * v.w);
}

// ---------------------------------------------------------------------------
// Cluster pooling
// ---------------------------------------------------------------------------
__global__ void cluster_init_kernel(float* __restrict__ csum,
                                    float* __restrict__ ccnt) {
  const int i = blockIdx.x * blockDim.x + threadIdx.x;
  if (i >= kC * kD) return;
  csum[i] = 0.f;
  if ((i & (kD - 1)) == 0) ccnt[i >> 7] = 0.f;
}

__global__ __launch_bounds__(256) void cluster_acc_kernel(
    const float* __restrict__ emb, const int* __restrict__ cl,
    float* __restrict__ csum, float* __restrict__ ccnt) {
  const int lane = threadIdx.x & 31;
  const int n = blockIdx.x * 8 + (threadIdx.x >> 5);
  if (n >= kN) return;
  const int c = cl[n];
  const float4 v = *(const float4*)&emb[(size_t)n * kD + lane * 4];
  float* o = csum + (size_t)c * kD + lane * 4;
  atomicAdd(o + 0, v.x);
  atomicAdd(o + 1, v.y);
  atomicAdd(o + 2, v.z);
  atomicAdd(o + 3, v.w);
  if (lane == 0) atomicAdd(&ccnt[c], 1.f);
}

__global__ void cluster_div_kernel(const float* __restrict__ csum,
                                   const float* __restrict__ ccnt,
                                   float* __restrict__ t) {
  const int i = blockIdx.x * blockDim.x + threadIdx.x;
  if (i >= kC * kD) return;
  t[i] = csum[i] / fmaxf(ccnt[i >> 7], 1.f);
}

// ---------------------------------------------------------------------------
// Final classifier: out = t[0] @ cls_w + cls_b
// ---------------------------------------------------------------------------
__global__ void cls_kernel(const float* __restrict__ t,
                           const float* __restrict__ w,
                           const float* __restrict__ b,
                           float* __restrict__ out) {
  const int j = threadIdx.x;
  if (j >= kNCLS) return;
  float acc = b[j];
  for (int k = 0; k < kD; ++k) acc += t[k] * w[k * kNCLS + j];
  out[j] = acc;
}

// ---------------------------------------------------------------------------
// Host-side orchestration
// ---------------------------------------------------------------------------
extern "C" void kernel_launch(void* const* d_in, const int* in_sizes, int n_in,
                              void* d_out, int out_size, void* d_ws, size_t ws_size,
                              hipStream_t stream) {
  (void)in_sizes; (void)n_in; (void)out_size; (void)ws_size;
  const float* x        = (const float*)d_in[0];
  const float* input_w  = (const float*)d_in[1];
  const float* input_b  = (const float*)d_in[2];
  const float* gat1_w   = (const float*)d_in[3];
  const float* gat1_as  = (const float*)d_in[4];
  const float* gat1_ad  = (const float*)d_in[5];
  const float* gat1_b   = (const float*)d_in[6];
  const float* gat2_w   = (const float*)d_in[7];
  const float* gat2_as  = (const float*)d_in[8];
  const float* gat2_ad  = (const float*)d_in[9];
  const float* gat2_b   = (const float*)d_in[10];
  const float* emb_w    = (const float*)d_in[11];
  const float* emb_b    = (const float*)d_in[12];
  const float* ln1_g    = (const float*)d_in[13];
  const float* ln1_b    = (const float*)d_in[14];
  const float* qkv_w    = (const float*)d_in[15];
  const float* qkv_b    = (const float*)d_in[16];
  const float* proj_w   = (const float*)d_in[17];
  const float* proj_b   = (const float*)d_in[18];
  const float* ln2_g    = (const float*)d_in[19];
  const float* ln2_b    = (const float*)d_in[20];
  const float* ff1_w    = (const float*)d_in[21];
  const float* ff1_b    = (const float*)d_in[22];
  const float* ff2_w    = (const float*)d_in[23];
  const float* ff2_b    = (const float*)d_in[24];
  const float* cls_w    = (const float*)d_in[25];
  const float* cls_b    = (const float*)d_in[26];
  const int* edge_index = (const int*)d_in[27];
  const int* clusters   = (const int*)d_in[28];
  // d_in[29] = num_clusters (compile-time constant here)

  // Workspace layout (floats)
  float* ws = (float*)d_ws;
  float* bufA = ws;                 ws += (size_t)kN * kD;
  float* bufB = ws;                 ws += (size_t)kN * kD;
  float* bufC = ws;                 ws += (size_t)kN * kD;
  float* esb  = ws;                 ws += kN;
  float* edb  = ws;                 ws += kN;
  unsigned* menc = (unsigned*)ws;   ws += kN;
  float* zbuf = ws;                 ws += kN;
  float* csum = ws;                 ws += (size_t)kC * kD;
  float* ccnt = ws;                 ws += kC;
  float* tb   = ws;                 ws += (size_t)kS * kD;
  float* yb   = ws;                 ws += (size_t)kS * kD;
  float* qkvb = ws;                 ws += (size_t)kS * kTD;
  float* attb = ws;                 ws += (size_t)kS * kD;
  float* ffb  = ws;                 ws += (size_t)kS * kMLP;

  auto gemm = [&](const float* A, const float* W, const float* bias,
                  const float* res, float* out, int M, int Nn, int K, int act) {
    const int tiles  = (M / 16) * (Nn / 32);
    const int blocks = (tiles + 7) / 8;
    gemm_kernel<<<blocks, 256, 0, stream>>>(A, W, bias, res, out, M, Nn, K, act);
  };

  auto gat = [&](const float* hin, const float* W, const float* as,
                 const float* ad, const float* bias, float* hh, float* outb) {
    gemm(hin, W, nullptr, nullptr, hh, kN, kD, kD, 0);              // h = h_in @ W
    gat_dots_kernel<<<kN / 8, 256, 0, stream>>>(hh, as, ad, esb, edb);
    gat_init_kernel<<<(kN * kD + 255) / 256, 256, 0, stream>>>(outb, bias, menc, zbuf);
    edge_max_kernel<<<(kEdges + 255) / 256, 256, 0, stream>>>(esb, edb, edge_index, menc);
    edge_sum_kernel<<<(kEdges + 255) / 256, 256, 0, stream>>>(esb, edb, edge_index, menc, zbuf);
    edge_aggr_kernel<<<(kEdges + 7) / 8, 256, 0, stream>>>(esb, edb, edge_index,
                                                           menc, zbuf, hh, outb);
  };

  // 1) input projection
  gemm(x, input_w, input_b, nullptr, bufA, kN, kD, kF, 0);
  // 2) two GAT layers
  gat(bufA, gat1_w, gat1_as, gat1_ad, gat1_b, bufB, bufC);  // bufC = gat1 out
  gat(bufC, gat2_w, gat2_as, gat2_ad, gat2_b, bufA, bufB);  // bufB = gat2 out
  // 3) embedding + cluster mean pooling
  gemm(bufB, emb_w, emb_b, nullptr, bufA, kN, kD, kD, 0);
  cluster_init_kernel<<<(kC * kD + 255) / 256, 256, 0, stream>>>(csum, ccnt);
  cluster_acc_kernel<<<kN / 8, 256, 0, stream>>>(bufA, clusters, csum, ccnt);
  cluster_div_kernel<<<(kC * kD + 255) / 256, 256, 0, stream>>>(csum, ccnt, tb);

  // 4) transformer layers
  for (int i = 0; i < kL; ++i) {
    ln_kernel<<<kS / 8, 256, 0, stream>>>(tb, ln1_g + i * kD, ln1_b + i * kD, yb, kS);
    gemm(yb, qkv_w + (size_t)i * kD * kTD, qkv_b + (size_t)i * kTD, nullptr,
         qkvb, kS, kTD, kD, 0);
    attn_kernel<<<(kH * (kS / 16)) / 8, 256, 0, stream>>>(qkvb, attb);
    gemm(attb, proj_w + (size_t)i * kD * kD, proj_b + (size_t)i * kD, tb, tb,
         kS, kD, kD, 0);
    ln_kernel<<<kS / 8, 256, 0, stream>>>(tb, ln2_g + i * kD, ln2_b + i * kD, yb, kS);
    gemm(yb, ff1_w + (size_t)i * kD * kMLP, ff1_b + (size_t)i * kMLP, nullptr,
         ffb, kS, kMLP, kD, 1 /*gelu*/);
    gemm(ffb, ff2_w + (size_t)i * kMLP * kD, ff2_b + (size_t)i * kD, tb, tb,
         kS, kD, kMLP, 0);
  }

  // 5) classifier head on token 0
  cls_kernel<<<1, 32, 0, stream>>>(tb, cls_w, cls_b, (float*)d_out);
}